// GGANN_8074538517121
// MI455X (gfx1250) — compile-verified
//
#include <hip/hip_runtime.h>
#include <math.h>

#define NN 8192
#define FF 256
#define ALPHA 0.2f
#define NEG_INF_C (-9.0e15f)

typedef __attribute__((ext_vector_type(2))) float v2f;
typedef __attribute__((ext_vector_type(8))) float v8f;
typedef __attribute__((ext_vector_type(4))) unsigned int v4u;
typedef __attribute__((ext_vector_type(8))) int v8i;
typedef __attribute__((ext_vector_type(4))) int v4i;

// D = A(16x4 f32) * B(4x16 f32) + C(16x16 f32), wave32 WMMA
__device__ __forceinline__ v8f wmma4(v2f a, v2f b, v8f c) {
  return __builtin_amdgcn_wmma_f32_16x16x4_f32(
      /*neg_a=*/false, a, /*neg_b=*/false, b,
      /*c_mod=*/(short)0, c, /*reuse_a=*/false, /*reuse_b=*/false);
}

// TDM: DMA a 16-row x 256-col f32 tile of Wh (rows jb..jb+15) into LDS.
// D# group0: count=1 | lds_addr | global_addr(57b) | type=2
// D# group1: data_size=4B, tensor_dim0=256, tensor_dim1=8192,
//            tile_dim0=256, tile_dim1=16, tensor_dim0_stride=256
__device__ __forceinline__ void tdm_load_wh_tile(const float* Wh, int jb,
                                                 unsigned int ldsOff) {
  unsigned long long ga = (unsigned long long)(size_t)(Wh + (size_t)jb * FF);
  v4u g0 = {1u,                                     // count=1 (user mode)
            ldsOff,                                 // lds_addr (bytes)
            (unsigned int)ga,                       // global_addr[31:0]
            ((unsigned int)(ga >> 32) & 0x1FFFFFFu) // global_addr[56:32]
                | 0x80000000u};                     // type=2 ("image")
  v8i g1 = {(int)0x00020000u,        // wg_mask=0, data_size=2 (4B)
            (int)(256u << 16),       // tensor_dim0[15:0]=256   (bits 79:48)
            (int)(8192u << 16),      // tensor_dim1[15:0]=8192  (bits 111:80)
            (int)(256u << 16),       // tile_dim0=256           (bits 127:112)
            16,                      // tile_dim1=16, tile_dim2=0
            256,                     // tensor_dim0_stride[31:0]=256
            0, 0};
  v4i gz = {0, 0, 0, 0};
#if __clang_major__ >= 23
  v8i gz8 = {0, 0, 0, 0, 0, 0, 0, 0};
  __builtin_amdgcn_tensor_load_to_lds(g0, g1, gz, gz, gz8, 0);
#else
  __builtin_amdgcn_tensor_load_to_lds(g0, g1, gz, gz, 0);
#endif
}

// ---------------------------------------------------------------------------
// Kernel 1: Wa1 = W @ a[:F], Wa2 = W @ a[F:]   (factorization of src/dst)
// ---------------------------------------------------------------------------
__global__ void k_wa(const float* __restrict__ W, const float* __restrict__ a,
                     float* __restrict__ Wa1, float* __restrict__ Wa2) {
  int k = threadIdx.x;  // 0..255
  float s1 = 0.f, s2 = 0.f;
  for (int f = 0; f < FF; ++f) {
    float w = W[k * FF + f];
    s1 += w * a[f];
    s2 += w * a[FF + f];
  }
  Wa1[k] = s1;
  Wa2[k] = s2;
}

// ---------------------------------------------------------------------------
// Kernel 2: src[i] = h_i . Wa1 ; dst[i] = h_i . Wa2   (one wave per row)
// ---------------------------------------------------------------------------
__global__ void k_srcdst(const float* __restrict__ h, const float* __restrict__ Wa1,
                         const float* __restrict__ Wa2, float* __restrict__ src,
                         float* __restrict__ dst) {
  int row  = (blockIdx.x * blockDim.x + threadIdx.x) >> 5;
  int lane = threadIdx.x & 31;
  const float* hr = h + (size_t)row * FF;
  float s1 = 0.f, s2 = 0.f;
#pragma unroll
  for (int t = 0; t < 8; ++t) {
    int f = lane + 32 * t;
    float hv = hr[f];
    s1 += hv * Wa1[f];
    s2 += hv * Wa2[f];
  }
#pragma unroll
  for (int m = 16; m >= 1; m >>= 1) {
    s1 += __shfl_xor(s1, m, 32);
    s2 += __shfl_xor(s2, m, 32);
  }
  if (lane == 0) { src[row] = s1; dst[row] = s2; }
}

// ---------------------------------------------------------------------------
// Kernel 3: Wh = h @ W via f32 WMMA. One wave per 16x16 output tile.
// ---------------------------------------------------------------------------
__global__ void __launch_bounds__(256) k_wh(const float* __restrict__ h,
                                            const float* __restrict__ W,
                                            float* __restrict__ Wh) {
  int wave = (blockIdx.x * blockDim.x + threadIdx.x) >> 5;
  int lane = threadIdx.x & 31;
  int half = lane >> 4, lm = lane & 15;
  int rowBase = (wave >> 4) * 16;  // 512 row tiles
  int colBase = (wave & 15) * 16;  // 16  col tiles
  v8f acc = {0.f, 0.f, 0.f, 0.f, 0.f, 0.f, 0.f, 0.f};
  const float2* hA = (const float2*)(h + (size_t)(rowBase + lm) * FF);
  for (int k0 = 0; k0 < FF; k0 += 4) {
    int ka = k0 + 2 * half;                 // K slot for this lane half
    float2 av = hA[ka >> 1];                // A[row=lm][ka, ka+1]
    v2f A; A.x = av.x; A.y = av.y;
    v2f B;                                  // B[k][n] = W[k][n]
    B.x = W[(size_t)ka * FF + colBase + lm];
    B.y = W[(size_t)(ka + 1) * FF + colBase + lm];
    acc = wmma4(A, B, acc);
  }
#pragma unroll
  for (int r = 0; r < 8; ++r)
    Wh[(size_t)(rowBase + r + 8 * half) * FF + colBase + lm] = acc[r];
}

// ---------------------------------------------------------------------------
// Kernel 4: flash-attention pass.  One wave owns 16 rows x all 256 cols.
// adj streamed exactly once (+prefetch); online softmax; P@Wh via WMMA.
// Wh tiles double-buffered in LDS via the Tensor Data Mover (wave 0 issues,
// s_wait_tensorcnt + workgroup barrier hand the tile to all 8 waves).
// ---------------------------------------------------------------------------
__global__ void __launch_bounds__(256) k_attn(const float* __restrict__ adj,
                                              const float* __restrict__ Wh,
                                              const float* __restrict__ src,
                                              const float* __restrict__ dst,
                                              float* __restrict__ hprime) {
  __shared__ float whT[2][16 * FF];  // 2 x 16 KB double buffer
  int waveInBlk = threadIdx.x >> 5;
  int lane = threadIdx.x & 31;
  int half = lane >> 4, lm = lane & 15;
  int rowBase = (blockIdx.x * 8 + waveInBlk) * 16;  // 512 row-blocks total

  unsigned int ldsOff[2];
  ldsOff[0] = (unsigned int)(size_t)(&whT[0][0]);  // low 32 bits = LDS offset
  ldsOff[1] = (unsigned int)(size_t)(&whT[1][0]);

  float srci = src[rowBase + lm];  // this lane's attention row = rowBase+lm
  const float* adjRow = adj + (size_t)(rowBase + lm) * NN;

  float m = -INFINITY, lsum = 0.f;
  v8f acc[16];
  v8f zero = {0.f, 0.f, 0.f, 0.f, 0.f, 0.f, 0.f, 0.f};
#pragma unroll
  for (int t = 0; t < 16; ++t) acc[t] = zero;

  // prologue: kick off tile 0
  if (waveInBlk == 0) tdm_load_wh_tile(Wh, 0, ldsOff[0]);

  for (int jb = 0; jb < NN; jb += 16) {
    int cur = (jb >> 4) & 1;
    // tile jb landed (no-op for waves that issued nothing)
    __builtin_amdgcn_s_wait_tensorcnt(0);
    __syncthreads();  // all waves: tile visible AND other buffer free
    if (waveInBlk == 0 && jb + 16 < NN)
      tdm_load_wh_tile(Wh, jb + 16, ldsOff[cur ^ 1]);  // overlap next tile
    const float* wt = &whT[cur][0];

    // keep the HBM-bound adj stream ahead of the VALU work
    if (jb + 64 < NN)
      __builtin_prefetch(adjRow + jb + 64 + 2 * half, 0, 0);

    // --- masked logits directly in the WMMA A-operand layout ---
    // pair q covers K=4q..4q+3; lane holds K = 4q+2*half, +1
    float p[8];
    float tmax = -INFINITY;
#pragma unroll
    for (int q = 0; q < 4; ++q) {
      int j0 = jb + 4 * q + 2 * half;
      float2 dj = *(const float2*)(dst + j0);
      float2 aj = *(const float2*)(adjRow + j0);
      float e0 = srci + dj.x; e0 = (e0 > 0.f) ? e0 : ALPHA * e0;
      float e1 = srci + dj.y; e1 = (e1 > 0.f) ? e1 : ALPHA * e1;
      e0 = (aj.x > 0.f) ? e0 : NEG_INF_C;
      e1 = (aj.y > 0.f) ? e1 : NEG_INF_C;
      p[2 * q] = e0; p[2 * q + 1] = e1;
      tmax = fmaxf(tmax, fmaxf(e0, e1));
    }
    // combine with partner lane (same row, other K half)
    tmax = fmaxf(tmax, __shfl_xor(tmax, 16, 32));
    float m_new = fmaxf(m, tmax);
    float corr = __expf(m - m_new);  // first tile: exp(-inf)=0
    float rsum = 0.f;
#pragma unroll
    for (int k = 0; k < 8; ++k) { p[k] = __expf(p[k] - m_new); rsum += p[k]; }
    rsum += __shfl_xor(rsum, 16, 32);
    lsum = lsum * corr + rsum;
    m = m_new;

    // rescale accumulators: row (r+8*half) scale lives at lane (r+8*half)
    float cr[8];
#pragma unroll
    for (int r = 0; r < 8; ++r) cr[r] = __shfl(corr, r + 8 * half, 32);
#pragma unroll
    for (int t = 0; t < 16; ++t) {
#pragma unroll
      for (int r = 0; r < 8; ++r) acc[t][r] *= cr[r];
    }

    // acc[t] += P(16x16) @ Wh_tile(16x16cols), K split into 4 WMMAs
#pragma unroll
    for (int t = 0; t < 16; ++t) {
#pragma unroll
      for (int q = 0; q < 4; ++q) {
        int kk = 4 * q + 2 * half;
        v2f A; A.x = p[2 * q]; A.y = p[2 * q + 1];
        v2f B;
        B.x = wt[kk * FF + t * 16 + lm];
        B.y = wt[(kk + 1) * FF + t * 16 + lm];
        acc[t] = wmma4(A, B, acc[t]);
      }
    }
  }

  // normalize by row sums and store
  float inv[8];
#pragma unroll
  for (int r = 0; r < 8; ++r) inv[r] = 1.0f / __shfl(lsum, r + 8 * half, 32);
#pragma unroll
  for (int t = 0; t < 16; ++t) {
#pragma unroll
    for (int r = 0; r < 8; ++r)
      hprime[(size_t)(rowBase + r + 8 * half) * FF + t * 16 + lm] =
          acc[t][r] * inv[r];
  }
}

// ---------------------------------------------------------------------------
// Kernel 5: fused GRU cell. One wave per 16x16 output tile; 6 WMMA
// accumulators (r,z,n gates for both h_prime@w_ih^T and h@w_hh^T).
// ---------------------------------------------------------------------------
__global__ void __launch_bounds__(256) k_gru(const float* __restrict__ hprime,
                                             const float* __restrict__ h,
                                             const float* __restrict__ wih,
                                             const float* __restrict__ whh,
                                             const float* __restrict__ bih,
                                             const float* __restrict__ bhh,
                                             float* __restrict__ out) {
  int wave = (blockIdx.x * blockDim.x + threadIdx.x) >> 5;
  int lane = threadIdx.x & 31;
  int half = lane >> 4, lm = lane & 15;
  int rowBase = (wave >> 4) * 16;
  int colBase = (wave & 15) * 16;

  v8f zero = {0.f, 0.f, 0.f, 0.f, 0.f, 0.f, 0.f, 0.f};
  v8f gi[3], gh[3];
#pragma unroll
  for (int g = 0; g < 3; ++g) { gi[g] = zero; gh[g] = zero; }

  const float2* Ai = (const float2*)(hprime + (size_t)(rowBase + lm) * FF);
  const float2* Ah = (const float2*)(h + (size_t)(rowBase + lm) * FF);
  for (int k0 = 0; k0 < FF; k0 += 4) {
    int ka = k0 + 2 * half;
    float2 a1 = Ai[ka >> 1];
    float2 a2 = Ah[ka >> 1];
    v2f A1; A1.x = a1.x; A1.y = a1.y;
    v2f A2; A2.x = a2.x; A2.y = a2.y;
#pragma unroll
    for (int g = 0; g < 3; ++g) {
      // B[k][n] = w[(g*F + colBase+n)][k]  -> K-contiguous float2 loads
      float2 b1 = *(const float2*)(wih + (size_t)(g * FF + colBase + lm) * FF + ka);
      v2f B1; B1.x = b1.x; B1.y = b1.y;
      gi[g] = wmma4(A1, B1, gi[g]);
      float2 b2 = *(const float2*)(whh + (size_t)(g * FF + colBase + lm) * FF + ka);
      v2f B2; B2.x = b2.x; B2.y = b2.y;
      gh[g] = wmma4(A2, B2, gh[g]);
    }
  }

#pragma unroll
  for (int r = 0; r < 8; ++r) {
    int row = rowBase + r + 8 * half;
    int col = colBase + lm;
    float ir = gi[0][r] + bih[col];
    float iz = gi[1][r] + bih[FF + col];
    float in_ = gi[2][r] + bih[2 * FF + col];
    float hr = gh[0][r] + bhh[col];
    float hz = gh[1][r] + bhh[FF + col];
    float hn = gh[2][r] + bhh[2 * FF + col];
    float rg = 1.f / (1.f + __expf(-(ir + hr)));
    float zg = 1.f / (1.f + __expf(-(iz + hz)));
    float ng = tanhf(in_ + rg * hn);
    float hv = h[(size_t)row * FF + col];
    out[(size_t)row * FF + col] = (1.f - zg) * ng + zg * hv;
  }
}

// ---------------------------------------------------------------------------
extern "C" void kernel_launch(void* const* d_in, const int* in_sizes, int n_in,
                              void* d_out, int out_size, void* d_ws, size_t ws_size,
                              hipStream_t stream) {
  const float* h   = (const float*)d_in[0];
  const float* adj = (const float*)d_in[1];
  const float* W   = (const float*)d_in[2];
  const float* a   = (const float*)d_in[3];
  const float* wih = (const float*)d_in[4];
  const float* whh = (const float*)d_in[5];
  const float* bih = (const float*)d_in[6];
  const float* bhh = (const float*)d_in[7];
  float* out = (float*)d_out;

  float* ws  = (float*)d_ws;                 // ~16.8 MB of fp32 scratch
  float* Wh  = ws;                           // [N,F]
  float* hp  = Wh + (size_t)NN * FF;         // [N,F]
  float* src = hp + (size_t)NN * FF;         // [N]
  float* dst = src + NN;                     // [N]
  float* Wa1 = dst + NN;                     // [F]
  float* Wa2 = Wa1 + FF;                     // [F]

  k_wa<<<1, 256, 0, stream>>>(W, a, Wa1, Wa2);
  k_srcdst<<<NN / 8, 256, 0, stream>>>(h, Wa1, Wa2, src, dst);
  k_wh<<<(NN / 16) * (FF / 16) / 8, 256, 0, stream>>>(h, W, Wh);
  k_attn<<<NN / 16 / 8, 256, 0, stream>>>(adj, Wh, src, dst, hp);
  k_gru<<<(NN / 16) * (FF / 16) / 8, 256, 0, stream>>>(hp, h, wih, whh, bih, bhh, out);
}